// dsq_loss_71914932404669
// MI455X (gfx1250) — compile-verified
//
#include <hip/hip_runtime.h>
#include <math.h>

typedef __attribute__((ext_vector_type(2))) float v2f;
typedef __attribute__((ext_vector_type(8))) float v8f;

#define NSIDE    128
#define KBINS    100
#define BATCH    8
#define LDSPITCH 132   // 128 + pad; row stride 528B (16B aligned), breaks bank conflicts

// ---- compile-time constants reproducing the reference binning ----
constexpr double PI_D   = 3.14159265358979323846;
constexpr double KFU_D  = 2.0 * PI_D / (NSIDE * 3.0);          // 2*pi*fftfreq step
constexpr double KMAX_D = KFU_D * 64.0 * 1.7320508075688772;   // kfu*64*sqrt(3)
constexpr float  KFU    = (float)KFU_D;
constexpr float  E0F    = (float)KFU_D;                        // min positive |k|
constexpr float  DELTAF = (float)((KMAX_D - KFU_D) / (double)KBINS);
constexpr float  INVD   = (float)((double)KBINS / (KMAX_D - KFU_D));
constexpr float  SCALEF = (float)(27.0 / 2097152.0);           // Lpix^3 / N^3
constexpr float  TWO_PI2 = (float)(2.0 * PI_D * PI_D);

#define BUFSZ       16777216u          // 8*128^3 floats (one complex component)
#define OFF_FRE     0u
#define OFF_FIM     16384u
#define OFF_CNT     32768u             // 128 ints
#define OFF_SUMS    32896u             // 2*8*100 floats
#define OFF_PART    34560u             // 2 fields * 8 b * 1024 chunks * 100 bins
#define PART_FIELD  (BATCH * 1024 * KBINS)    // 819200
#define OFF_BUF     2097152u           // 1<<21 floats, then 4 component buffers

// ------------------------------------------------------------------
__global__ void zero_counts_kernel(int* counts) {
    int i = threadIdx.x;
    if (i < 128) counts[i] = 0;
}

// F[n*128+k] = exp(-2*pi*i*n*k/128); symmetric, stored row-major
__global__ void dft_init_kernel(float* __restrict__ Fre, float* __restrict__ Fim) {
    int i = blockIdx.x * 256 + threadIdx.x;     // 0..16383
    int n = i >> 7, k = i & 127;
    int t = (n * k) & 127;                      // exact phase mod N
    float th = (float)t * (float)(2.0 * PI_D / 128.0);
    float s, c;
    sincosf(th, &s, &c);
    Fre[i] = c;
    Fim[i] = -s;
}

__device__ __forceinline__ int freq_m(int i) { return ((i + 64) & 127) - 64; }

__device__ __forceinline__ int bin_of(int s) {  // s = integer |m|^2 > 0
    float km = KFU * sqrtf((float)s);
    int b = (int)floorf((km - E0F) * INVD);
    return min(max(b, 0), KBINS - 1);
}

// mode counts per bin over one 128^3 grid (DC dropped)
__global__ __launch_bounds__(256) void counts_kernel(int* __restrict__ counts) {
    __shared__ int lb[KBINS];
    for (int j = threadIdx.x; j < KBINS; j += 256) lb[j] = 0;
    __syncthreads();
    int base = blockIdx.x * 8192;
    for (int i = threadIdx.x; i < 8192; i += 256) {
        int r  = base + i;
        int m3 = freq_m(r >> 14), m2 = freq_m((r >> 7) & 127), m1 = freq_m(r & 127);
        int s  = m1 * m1 + m2 * m2 + m3 * m3;
        if (s) atomicAdd(&lb[bin_of(s)], 1);
    }
    __syncthreads();
    for (int j = threadIdx.x; j < KBINS; j += 256)
        if (lb[j]) atomicAdd(&counts[j], lb[j]);
}

// ------------------------------------------------------------------
// One DFT pass along the contiguous axis: Y[row][k] = sum_n X[row][n] * F[n][k]
// (F symmetric). Output axis-rotated so the next axis is contiguous:
//   out[b*2^21 + k*2^14 + (row & 16383)],  b = row >> 14.
// Block: 16 rows staged into LDS with GLOBAL_LOAD_ASYNC_TO_LDS_B128;
// 8 waves, wave w computes k in [16w,16w+16) via V_WMMA_F32_16X16X4_F32.
// FUSE_REDUCE (final pass): skip the store entirely and bin |Y|^2 into
// per-block partials (saves one full write+read of the 128MiB field).
// ------------------------------------------------------------------
template <bool REAL_IN, bool FUSE_REDUCE>
__global__ __launch_bounds__(256) void dft_pass_kernel(
    const float* __restrict__ in_re, const float* __restrict__ in_im,
    const float* __restrict__ Fre,   const float* __restrict__ Fim,
    float* __restrict__ out_re,      float* __restrict__ out_im,
    float* __restrict__ part) {

    __shared__ float X[2][16][LDSPITCH];
    __shared__ float lb[KBINS];

    const int tid     = threadIdx.x;
    const int rowBase = blockIdx.x << 4;

    if constexpr (FUSE_REDUCE) {
        for (int j = tid; j < KBINS; j += 256) lb[j] = 0.f;
    }

    // async stage: 16 rows x 128 cols per component, b128 direct to LDS
    for (int f = tid; f < 512; f += 256) {
        int r = f >> 5;              // 0..15
        int c = (f & 31) << 2;       // 0,4,...,124
        unsigned long long g0 = (unsigned long long)(in_re + (size_t)(rowBase + r) * 128 + c);
        unsigned           l0 = (unsigned)(uintptr_t)&X[0][r][c];
        asm volatile("global_load_async_to_lds_b128 %0, %1, off"
                     :: "v"(l0), "v"(g0) : "memory");
        if constexpr (!REAL_IN) {
            unsigned long long g1 = (unsigned long long)(in_im + (size_t)(rowBase + r) * 128 + c);
            unsigned           l1 = (unsigned)(uintptr_t)&X[1][r][c];
            asm volatile("global_load_async_to_lds_b128 %0, %1, off"
                         :: "v"(l1), "v"(g1) : "memory");
        }
    }
    asm volatile("s_wait_asynccnt 0" ::: "memory");
    __syncthreads();

    const int lane = tid & 31;
    const int wave = tid >> 5;
    const int h    = lane >> 4;          // lane half
    const int m    = lane & 15;          // A-matrix row (and B column id)
    const int kcol = (wave << 4) + m;    // output frequency handled by this lane

    v8f P = {}, Q = {}, R = {}, S = {};

    for (int s = 0; s < 32; ++s) {
        int n0 = (s << 2) + (h << 1);    // K pair for this lane half
        v2f a_re = *reinterpret_cast<const v2f*>(&X[0][m][n0]);
        // B fragment via symmetry: F[n][kcol] == F[kcol][n] -> contiguous b64
        v2f b_re = *reinterpret_cast<const v2f*>(Fre + kcol * 128 + n0);
        v2f b_im = *reinterpret_cast<const v2f*>(Fim + kcol * 128 + n0);
        P = __builtin_amdgcn_wmma_f32_16x16x4_f32(false, a_re, false, b_re, (short)0, P, false, false);
        R = __builtin_amdgcn_wmma_f32_16x16x4_f32(false, a_re, false, b_im, (short)0, R, false, false);
        if constexpr (!REAL_IN) {
            v2f a_im = *reinterpret_cast<const v2f*>(&X[1][m][n0]);
            Q = __builtin_amdgcn_wmma_f32_16x16x4_f32(false, a_im, false, b_im, (short)0, Q, false, false);
            S = __builtin_amdgcn_wmma_f32_16x16x4_f32(false, a_im, false, b_re, (short)0, S, false, false);
        }
    }

    v8f Yre, Yim;
    if constexpr (REAL_IN) { Yre = P;     Yim = R;     }
    else                   { Yre = P - Q; Yim = R + S; }

    if constexpr (FUSE_REDUCE) {
        // Final pass: bin |Y|^2 directly. kcol is the freq index of this pass's
        // axis; the output row (rowBase + j + 8h) & 16383 encodes (k2*128 + k1).
        const int m3 = freq_m(kcol);
        const int s3 = m3 * m3;
        #pragma unroll
        for (int j = 0; j < 8; ++j) {
            int row = rowBase + j + (h << 3);
            int m2  = freq_m((row >> 7) & 127);
            int m1  = freq_m(row & 127);
            int s   = m1 * m1 + m2 * m2 + s3;
            if (s) {
                float pk = Yre[j] * Yre[j] + Yim[j] * Yim[j];
                atomicAdd(&lb[bin_of(s)], pk);
            }
        }
        __syncthreads();
        int bb    = rowBase >> 14;
        int chunk = (rowBase & 16383) >> 4;          // 0..1023 within batch
        float* dst = part + ((size_t)(bb << 10) + chunk) * KBINS;
        for (int j = tid; j < KBINS; j += 256) dst[j] = lb[j];
    } else {
        // D layout: VGPR j <-> output row (j + 8h); transpose-on-store rotation
        size_t bb    = (size_t)(rowBase >> 14);
        size_t obase = (bb << 21) + ((size_t)kcol << 14) + (size_t)(rowBase & 16383) + ((size_t)h << 3);
        *reinterpret_cast<float4*>(out_re + obase)     = make_float4(Yre[0], Yre[1], Yre[2], Yre[3]);
        *reinterpret_cast<float4*>(out_re + obase + 4) = make_float4(Yre[4], Yre[5], Yre[6], Yre[7]);
        *reinterpret_cast<float4*>(out_im + obase)     = make_float4(Yim[0], Yim[1], Yim[2], Yim[3]);
        *reinterpret_cast<float4*>(out_im + obase + 4) = make_float4(Yim[4], Yim[5], Yim[6], Yim[7]);
    }
}

// ------------------------------------------------------------------
// Deterministic reduction of per-block partials: sums[f][b][bin]
// ------------------------------------------------------------------
__global__ __launch_bounds__(256) void chunk_sum_kernel(const float* __restrict__ part,
                                                        float* __restrict__ sums) {
    int t = blockIdx.x * 256 + threadIdx.x;
    if (t >= 2 * BATCH * KBINS) return;
    int f   = t / (BATCH * KBINS);
    int r   = t % (BATCH * KBINS);
    int b   = r / KBINS;
    int bin = r % KBINS;
    const float* p = part + (size_t)f * PART_FIELD + ((size_t)b << 10) * KBINS + bin;
    float s = 0.f;
    for (int c = 0; c < 1024; ++c) s += p[(size_t)c * KBINS];
    sums[t] = s;
}

__global__ __launch_bounds__(256) void final_kernel(const float* __restrict__ sums,
                                                    const int* __restrict__ counts,
                                                    float* __restrict__ out) {
    __shared__ float ssum[256];
    __shared__ int   scnt[256];
    float acc = 0.f; int an = 0;
    for (int e = threadIdx.x; e < BATCH * KBINS; e += 256) {
        int bin = e % KBINS;
        float sp = sums[e];                       // pred   (field 0)
        float st = sums[BATCH * KBINS + e];       // target (field 1)
        int cnt = counts[bin];
        float d = NAN;
        if (cnt > 0) {
            float kc  = E0F + ((float)bin + 0.5f) * DELTAF;
            float fac = SCALEF * kc * kc * kc / (TWO_PI2 * (float)cnt);
            d = fabsf(log10f(st * fac) - log10f(sp * fac));
        }
        if (!isnan(d)) { acc += d; an += 1; }
    }
    ssum[threadIdx.x] = acc; scnt[threadIdx.x] = an;
    __syncthreads();
    for (int off = 128; off; off >>= 1) {
        if (threadIdx.x < off) {
            ssum[threadIdx.x] += ssum[threadIdx.x + off];
            scnt[threadIdx.x] += scnt[threadIdx.x + off];
        }
        __syncthreads();
    }
    if (threadIdx.x == 0) out[0] = ssum[0] / (float)scnt[0];   // LOSS_WEIGHT == 1
}

// ------------------------------------------------------------------
extern "C" void kernel_launch(void* const* d_in, const int* in_sizes, int n_in,
                              void* d_out, int out_size, void* d_ws, size_t ws_size,
                              hipStream_t stream) {
    (void)in_sizes; (void)n_in; (void)out_size; (void)ws_size;
    const float* pred   = (const float*)d_in[0];
    const float* target = (const float*)d_in[1];
    float* ws  = (float*)d_ws;

    float* Fre  = ws + OFF_FRE;
    float* Fim  = ws + OFF_FIM;
    int*   cnts = (int*)(ws + OFF_CNT);
    float* sums = ws + OFF_SUMS;
    float* part = ws + OFF_PART;
    float* Are  = ws + OFF_BUF;
    float* Aim  = Are + BUFSZ;
    float* Bre  = Aim + BUFSZ;
    float* Bim  = Bre + BUFSZ;

    zero_counts_kernel<<<1, 128, 0, stream>>>(cnts);
    dft_init_kernel<<<64, 256, 0, stream>>>(Fre, Fim);
    counts_kernel<<<256, 256, 0, stream>>>(cnts);

    const float* fields[2] = {pred, target};
    for (int f = 0; f < 2; ++f) {
        dft_pass_kernel<true,  false><<<8192, 256, 0, stream>>>(
            fields[f], nullptr, Fre, Fim, Are, Aim, nullptr);
        dft_pass_kernel<false, false><<<8192, 256, 0, stream>>>(
            Are, Aim, Fre, Fim, Bre, Bim, nullptr);
        dft_pass_kernel<false, true ><<<8192, 256, 0, stream>>>(
            Bre, Bim, Fre, Fim, nullptr, nullptr, part + (size_t)f * PART_FIELD);
    }
    chunk_sum_kernel<<<7, 256, 0, stream>>>(part, sums);
    final_kernel<<<1, 256, 0, stream>>>(sums, cnts, (float*)d_out);
}